// LoNet_16655883173987
// MI455X (gfx1250) — compile-verified
//
#include <hip/hip_runtime.h>
#include <cstdint>
#include <math.h>

// ---------------------------------------------------------------------------
// T = prod_k Givens2x2(row r_k) @ I (complex64, n=384), then diag(exp(i*g)).
// Layered schedule: rotation (i,j) -> layer t=i+j; rotations within a layer
// are row-disjoint (r=(t&1)+2s) and layer order is a topological order of
// the sequential scan's dependency DAG. Depth: 765 layers vs 73,536 steps.
//
// Latency-bound problem (0.79 GFLOP, 1.18 MB out): independent 8-column
// slabs per workgroup, all 384 rows in LDS, one barrier per layer. Each
// layer's 3KB parameter table is streamed into LDS one layer ahead via the
// Tensor Data Mover (TENSORcnt) -- or the async-to-LDS pipe (ASYNCcnt) if
// the TDM builtin is unavailable. WMMA is intentionally not used: a banded
// group-product GEMM reformulation costs ~7x flops, its band-product build
// is as expensive as direct application, and nothing shortens the 765-layer
// critical path.
// ---------------------------------------------------------------------------

#define N_MODES   384
#define LAYERS    (2 * N_MODES - 3)        // 765 layers (t = 0 .. 764)
#define MAXW      (N_MODES / 2)            // 192 = max rotations per layer
#define COLS_WG   8                        // columns per workgroup (slab)
#define THREADS2  256                      // 8 wave32 waves
#define SGROUPS   (THREADS2 / COLS_WG)     // 32 slot-groups
#define ROW_PITCH (COLS_WG + 1)            // LDS pad (float2 units)
#define PRM_BYTES ((size_t)LAYERS * MAXW * 16)   // 2,350,080 B

#if defined(__has_builtin)
#  if __has_builtin(__builtin_amdgcn_tensor_load_to_lds)
#    define LONET_HAVE_TDM 1
#  else
#    define LONET_HAVE_TDM 0
#  endif
#else
#  define LONET_HAVE_TDM 0
#endif

// ---------------------------------------------------------------------------
// Per-slot parameter expansion: layer t, slot s -> i = ceil(t/2)+s, j = t-i,
// flat k = i(i+1)/2 + j;  params = {cos a, sin a, cos b, sin b}.
// ---------------------------------------------------------------------------
__device__ __forceinline__ float4 rot_params(const float* __restrict__ alphas,
                                             const float* __restrict__ betas,
                                             int t, int s) {
    int lo = (t + 1) >> 1;
    int hi = min(N_MODES - 2, t);
    int i  = lo + s;
    if (i <= hi) {
        int j = t - i;
        int k = (i * (i + 1)) / 2 + j;
        float sa, ca, sb, cb;
        sincosf(alphas[k], &sa, &ca);
        sincosf(betas[k],  &sb, &cb);
        return make_float4(ca, sa, cb, sb);
    }
    return make_float4(1.0f, 0.0f, 1.0f, 0.0f);     // identity (never applied)
}

// Kernel 1: expand (alpha, beta) into packed per-layer tables in d_ws.
__global__ __launch_bounds__(256)
void lonet_build_params(const float* __restrict__ alphas,
                        const float* __restrict__ betas,
                        float4* __restrict__ prm) {
    int idx = blockIdx.x * blockDim.x + threadIdx.x;
    if (idx >= LAYERS * MAXW) return;
    int t = idx / MAXW;
    int s = idx - t * MAXW;
    prm[idx] = rot_params(alphas, betas, t, s);
}

// ---------------------------------------------------------------------------
// Layer-table staging: TDM (primary) or async-to-LDS (fallback).
// ---------------------------------------------------------------------------
#if LONET_HAVE_TDM
typedef __attribute__((ext_vector_type(4))) unsigned int v4u;
typedef __attribute__((ext_vector_type(4))) int          v4i;
typedef __attribute__((ext_vector_type(8))) int          v8i;

// One 1-row tile (384 x 8B = 3KB) of the [LAYERS x 384] 8B-element tensor.
__device__ __forceinline__ void stage_issue(const float4* gsrc, float4* ldst, int tid) {
    if (tid < 32) {                                      // wave 0 issues TDM
        unsigned long long ga = (unsigned long long)(uintptr_t)gsrc;
        v4u g0;
        g0.x = 1u;                                       // count=1, user mode
        g0.y = (unsigned)(uintptr_t)ldst;                // lds_addr (bytes)
        g0.z = (unsigned)ga;                             // global_addr[31:0]
        g0.w = ((unsigned)(ga >> 32) & 0x01FFFFFFu)      // global_addr[56:32]
               | 0x80000000u;                            // type=2 ("image")
        v8i g1;
        g1[0] = 0x00030000;                              // data_size=3 (8B)
        g1[1] = (int)(((unsigned)(2 * MAXW) & 0xFFFFu) << 16);  // tdim0 lo16
        g1[2] = (int)(((unsigned)LAYERS     & 0xFFFFu) << 16);  // tdim1 lo16
        g1[3] = (int)(((unsigned)(2 * MAXW) & 0xFFFFu) << 16);  // tile_dim0=384
        g1[4] = 1;                                       // tile_dim1=1
        g1[5] = 2 * MAXW;                                // tdim0_stride=384
        g1[6] = (int)(((unsigned)(2 * MAXW) & 0xFFFFu) << 16);  // tdim1_stride
        g1[7] = 0;
        v4i z4 = {0, 0, 0, 0};                           // groups 2/3: <=2D
#if __has_include(<hip/amd_detail/amd_gfx1250_TDM.h>)
        v8i z8 = {0, 0, 0, 0, 0, 0, 0, 0};               // 6-arg toolchain
        __builtin_amdgcn_tensor_load_to_lds(g0, g1, z4, z4, z8, 0);
#else
        __builtin_amdgcn_tensor_load_to_lds(g0, g1, z4, z4, 0);  // ROCm 7.2
#endif
    }
}

__device__ __forceinline__ void stage_wait() {
#if __has_builtin(__builtin_amdgcn_s_wait_tensorcnt)
    __builtin_amdgcn_s_wait_tensorcnt(0);
#else
    asm volatile("s_wait_tensorcnt 0x0" ::: "memory");
#endif
}
#else   // ----- async-to-LDS fallback (ASYNCcnt) ----------------------------
__device__ __forceinline__ void stage_issue(const float4* gsrc, float4* ldst, int tid) {
    if (tid < MAXW) {
        uint32_t lds_addr = (uint32_t)(uintptr_t)(ldst + tid);
        unsigned long long gaddr = (unsigned long long)(uintptr_t)(gsrc + tid);
        asm volatile("global_load_async_to_lds_b128 %0, %1, off"
                     :: "v"(lds_addr), "v"(gaddr) : "memory");
    }
}

__device__ __forceinline__ void stage_wait() {
    asm volatile("s_wait_asynccnt 0" ::: "memory");
}
#endif

// ---------------------------------------------------------------------------
// Kernel 2: one workgroup owns an 8-column slab of T (all 384 rows in LDS),
// loops over 765 layers with one barrier each; next layer's table is staged
// into the other LDS buffer while the current layer's math runs.
//   PREFETCH=false: no-workspace fallback; params computed in-kernel.
// ---------------------------------------------------------------------------
template <bool PREFETCH>
__global__ __launch_bounds__(THREADS2)
void lonet_apply_rotations(const float4* __restrict__ prm,
                           const float* __restrict__ alphas,
                           const float* __restrict__ betas,
                           const float* __restrict__ gammas,
                           float2* __restrict__ out) {
    __shared__ float2 Ttile[N_MODES * ROW_PITCH];   // 384*9*8  = 27,648 B
    __shared__ float4 stage[2][MAXW];               // 2*192*16 =  6,144 B

    const int tid  = threadIdx.x;
    const int col  = tid & (COLS_WG - 1);
    const int sg   = tid >> 3;                      // slot-group 0..31
    const int gcol = blockIdx.x * COLS_WG + col;    // global column

    // --- T0 = identity restricted to this column slab -----------------------
    for (int r = sg; r < N_MODES; r += SGROUPS)
        Ttile[r * ROW_PITCH + col] = make_float2((r == gcol) ? 1.0f : 0.0f, 0.0f);

    if (PREFETCH) {
        stage_issue(prm, &stage[0][0], tid);        // layer 0 table
        stage_wait();
    }
    __syncthreads();

    int buf = 0;
    for (int t = 0; t < LAYERS; ++t) {
        if (PREFETCH) {
            if ((t + 1) < LAYERS)                   // stage layer t+1 tables
                stage_issue(prm + (size_t)(t + 1) * MAXW, &stage[buf ^ 1][0], tid);
        } else {
            if (tid < MAXW)                         // compute params in-kernel
                stage[0][tid] = rot_params(alphas, betas, t, tid);
            __syncthreads();
        }

        const int Lt     = min(N_MODES - 2, t) - ((t + 1) >> 1) + 1;
        const int base_r = t & 1;

        // apply this layer's Lt row-disjoint rotations to our 8 columns
        for (int s = sg; s < Lt; s += SGROUPS) {
            float4 p = stage[buf][s];               // {c, sin, e_re, e_im}
            const float c  = p.x, sn = p.y, er = p.z, ei = p.w;
            const int r = base_r + 2 * s;
            float2 x0 = Ttile[(r    ) * ROW_PITCH + col];
            float2 x1 = Ttile[(r + 1) * ROW_PITCH + col];
            // new0 =  c*x0 + conj(e)*s*x1 ; new1 = -e*s*x0 + c*x1
            float2 n0, n1;
            n0.x = fmaf(c, x0.x,  sn * fmaf(er, x1.x,  ei * x1.y));
            n0.y = fmaf(c, x0.y,  sn * fmaf(er, x1.y, -ei * x1.x));
            n1.x = fmaf(c, x1.x, -sn * fmaf(er, x0.x, -ei * x0.y));
            n1.y = fmaf(c, x1.y, -sn * fmaf(er, x0.y,  ei * x0.x));
            Ttile[(r    ) * ROW_PITCH + col] = n0;
            Ttile[(r + 1) * ROW_PITCH + col] = n1;
        }

        if (PREFETCH) {
            stage_wait();       // retire staged tables (TENSORcnt/ASYNCcnt)
            buf ^= 1;
        }
        __syncthreads();        // publish T updates + staged params
    }

    // --- epilogue: diag(exp(i*gamma)) @ T, store interleaved complex64 ------
    for (int r = sg; r < N_MODES; r += SGROUPS) {
        float gs, gc;
        sincosf(gammas[r], &gs, &gc);
        float2 v = Ttile[r * ROW_PITCH + col];
        float2 o;
        o.x = gc * v.x - gs * v.y;
        o.y = gc * v.y + gs * v.x;
        out[(size_t)r * N_MODES + gcol] = o;
    }
}

// ---------------------------------------------------------------------------
// Host launcher
//   d_in[0]=alphas (73,536 f32), d_in[1]=betas, d_in[2]=gammas (384 f32);
//   d_out = 384*384 complex64 (interleaved f32); d_ws = param tables.
// ---------------------------------------------------------------------------
extern "C" void kernel_launch(void* const* d_in, const int* in_sizes, int n_in,
                              void* d_out, int out_size, void* d_ws, size_t ws_size,
                              hipStream_t stream) {
    (void)in_sizes; (void)n_in; (void)out_size;
    const float* alphas = (const float*)d_in[0];
    const float* betas  = (const float*)d_in[1];
    const float* gammas = (const float*)d_in[2];
    float2* out = (float2*)d_out;

    const int nwg = N_MODES / COLS_WG;                       // 48 workgroups

    if (ws_size >= PRM_BYTES && d_ws != nullptr) {
        float4* prm = (float4*)d_ws;
        const int total = LAYERS * MAXW;                     // 146,880 slots
        lonet_build_params<<<(total + 255) / 256, 256, 0, stream>>>(alphas, betas, prm);
        lonet_apply_rotations<true><<<nwg, THREADS2, 0, stream>>>(
            prm, alphas, betas, gammas, out);
    } else {
        lonet_apply_rotations<false><<<nwg, THREADS2, 0, stream>>>(
            nullptr, alphas, betas, gammas, out);
    }
}